// SSVI_torch_47175920779589
// MI455X (gfx1250) — compile-verified
//
#include <hip/hip_runtime.h>
#include <math.h>

// ---------------- problem constants (from reference) ----------------
#define NDIM   3
#define NCOL   500000
#define RANK   64
#define K1     32
#define BATCH  8192
#define NUM_TRAIN 819200.0f

typedef __attribute__((ext_vector_type(2))) float v2f;
typedef __attribute__((ext_vector_type(8))) float v8f;

// Low 32 bits of a flat pointer into the LDS aperture == wave-relative LDS
// byte offset (CDNA5 ISA 10.2).
__device__ __forceinline__ unsigned lds_byte_off(const void* p) {
  return (unsigned)(unsigned long long)p;
}

// One workgroup (64 threads = 2 waves) per batch element b.
// LDS: eps slab [3][32][64] f32 (24KB) xor-swizzled at 16B-chunk granularity:
//   chunk (k, c) stored at chunk slot (c ^ (k & 15)) within row k.
// This keeps async b128 writes 16B-aligned and makes the WMMA-phase strided
// reads (lane -> row k, fixed chunk j) hit 16 distinct bank groups.
__global__ __launch_bounds__(64) void ssvi_main_kernel(
    const int*   __restrict__ entries,  // [B, NDIM]
    const float* __restrict__ ys,       // [B]
    const float* __restrict__ means,    // [NDIM, NCOL, RANK]
    const float* __restrict__ chols,    // [NDIM, NCOL, RANK]
    const float* __restrict__ eps,      // [NDIM, B, K1, RANK]
    float*       __restrict__ ws)       // [B] per-block partials
{
  __shared__ float s_eps[NDIM * K1 * RANK];  // 24 KB, swizzled
  __shared__ float s_m[NDIM * RANK];
  __shared__ float s_S[NDIM * RANK];
  __shared__ float s_kl[64];
  __shared__ float s_wsum[2];

  const int b    = blockIdx.x;
  const int tid  = threadIdx.x;   // 0..63
  const int lane = tid & 31;
  const int wave = tid >> 5;

  // ---- 1) issue async eps loads: HBM -> LDS, 16B per lane, coalesced ----
  const unsigned eps_base = lds_byte_off(s_eps);
  for (int d = 0; d < NDIM; ++d) {
    const float* src = eps + ((size_t)d * BATCH + (size_t)b) * (K1 * RANK);
    const unsigned dbase = eps_base + (unsigned)(d * K1 * RANK * 4);
    #pragma unroll
    for (int it = 0; it < 8; ++it) {
      const int n   = it * 64 + tid;     // linear 16B-chunk id, 0..511
      const int row = n >> 4;            // k
      const int ch  = n & 15;            // chunk within row
      const float* g = src + row * RANK + ch * 4;  // contiguous across lanes
      const unsigned l = dbase + (unsigned)(row * 256 + ((ch ^ (row & 15)) << 4));
      asm volatile("global_load_async_to_lds_b128 %0, %1, off"
                   :: "v"(l), "v"(g) : "memory");
    }
  }

  // ---- 2) gather mean/chol rows (coalesced 256B rows), KL partials ----
  float kl = 0.0f;
  #pragma unroll
  for (int d = 0; d < NDIM; ++d) {
    const int idx = entries[b * NDIM + d];
    const size_t base = ((size_t)d * NCOL + (size_t)idx) * RANK + tid;
    const float m = means[base];
    const float L = chols[base];
    const float S = L * L;                 // reference scale = L^2
    s_m[d * RANK + tid] = m;
    s_S[d * RANK + tid] = S;
    const float var_ratio = S * S;
    kl += 0.5f * (var_ratio + m * m - 1.0f - logf(var_ratio));
  }
  s_kl[tid] = kl;

  // async eps must be fully landed in LDS before either wave reads it
  asm volatile("s_wait_asynccnt 0" ::: "memory");
  __syncthreads();

  // ---- 3) WMMA row-sum: fs[k] = sum_r prod_d (m + eps * S) ----
  // Wave w owns k = 16w .. 16w+15.  A-matrix (16x4 f32) layout:
  //   lanes 0-15: M = lane, K-slots {4j, 4j+1}; lanes 16-31: M = lane-16,
  //   K-slots {4j+2, 4j+3}.  B = all-ones 4x16 => D[m, n] = row sum, so after
  //   accumulating j = 0..15, c[v] holds fs[kbase+v] (lanes 0-15) and
  //   fs[kbase+8+v] (lanes 16-31).
  const int kbase = wave * 16;
  const int k     = kbase + (lane & 15);
  const int half  = lane >> 4;
  v8f c = {};
  v2f onesb; onesb.x = 1.0f; onesb.y = 1.0f;
  #pragma unroll
  for (int j = 0; j < 16; ++j) {
    const int r0  = 4 * j + 2 * half;          // even => float2 is 8B aligned
    const int swz = (j ^ (k & 15)) << 2;       // swizzled chunk -> float index
    float px = 1.0f, py = 1.0f;
    #pragma unroll
    for (int d = 0; d < NDIM; ++d) {
      const float2 e = *(const float2*)&s_eps[d * (K1 * RANK) + k * RANK + swz + 2 * half];
      const float2 m2 = *(const float2*)&s_m[d * RANK + r0];
      const float2 S2 = *(const float2*)&s_S[d * RANK + r0];
      px *= fmaf(e.x, S2.x, m2.x);
      py *= fmaf(e.y, S2.y, m2.y);
    }
    v2f a; a.x = px; a.y = py;                 // a[0]=even K-slot, a[1]=odd
    c = __builtin_amdgcn_wmma_f32_16x16x4_f32(
        /*neg_a=*/false, a, /*neg_b=*/false, onesb,
        /*c_mod=*/(short)0, c, /*reuse_a=*/false, /*reuse_b=*/false);
  }

  // ---- 4) log-pdf over this wave's 16 samples ----
  const float y = ys[b];
  const float HALF_LOG2PI = 0.91893853320467274178f;
  float ss = 0.0f;
  #pragma unroll
  for (int v = 0; v < 8; ++v) {
    const float diff = c[v] - y;
    ss = fmaf(-0.5f * diff, diff, ss) - HALF_LOG2PI;
  }
  ss += __shfl_xor(ss, 16, 32);   // lanes 0-15 cover M=0-7, 16-31 cover M=8-15
  if (lane == 0) s_wsum[wave] = ss;
  __syncthreads();

  // ---- 5) per-block partial (deterministic; no float atomics) ----
  if (tid == 0) {
    float klsum = 0.0f;
    for (int i = 0; i < 64; ++i) klsum += s_kl[i];
    const float expect_b = (s_wsum[0] + s_wsum[1]) * (1.0f / (float)K1);
    ws[b] = (NUM_TRAIN / (float)BATCH) * expect_b - klsum / (float)BATCH;
  }
}

__global__ __launch_bounds__(256) void ssvi_reduce_kernel(
    const float* __restrict__ ws, float* __restrict__ out)
{
  __shared__ float red[256];
  float acc = 0.0f;
  for (int i = threadIdx.x; i < BATCH; i += 256) acc += ws[i];
  red[threadIdx.x] = acc;
  __syncthreads();
  for (int s = 128; s > 0; s >>= 1) {
    if (threadIdx.x < s) red[threadIdx.x] += red[threadIdx.x + s];
    __syncthreads();
  }
  if (threadIdx.x == 0) out[0] = -red[0];   // loss = -(sum of partials)
}

extern "C" void kernel_launch(void* const* d_in, const int* in_sizes, int n_in,
                              void* d_out, int out_size, void* d_ws, size_t ws_size,
                              hipStream_t stream) {
  (void)in_sizes; (void)n_in; (void)out_size; (void)ws_size;
  const int*   entries = (const int*)d_in[0];
  const float* ys      = (const float*)d_in[1];
  const float* means   = (const float*)d_in[2];
  const float* chols   = (const float*)d_in[3];
  const float* eps     = (const float*)d_in[4];
  float* ws = (float*)d_ws;                  // BATCH floats of scratch (32 KB)

  ssvi_main_kernel<<<BATCH, 64, 0, stream>>>(entries, ys, means, chols, eps, ws);
  ssvi_reduce_kernel<<<1, 256, 0, stream>>>(ws, (float*)d_out);
}